// TensorProduct_53180285059258
// MI455X (gfx1250) — compile-verified
//
#include <hip/hip_runtime.h>

typedef __attribute__((ext_vector_type(2))) float v2f;
typedef __attribute__((ext_vector_type(8))) float v8f;

#define L1DIM 9
#define L2DIM 9
#define NDIM 25      // (2+2+1)^2
#define KDIM 81      // 9*9 flattened (l,m)
#define FDIM 512
#define NK   21      // ceil(81/4) K-steps of the 16x16x4 WMMA
#define TILES_PER_WAVE 4

// out[b][p][n][f] = parity_even[p] * sum_{l,m} cg[l][m][n] * x1[b][l][f] * x2[b][m][f]
// GEMM view per (b, f-tile of 16): D[n][f] = sum_kk cgT[n][kk] * outer[kk][f]
__global__ __launch_bounds__(256) void cg_tp_wmma(
    const float* __restrict__ x1, const float* __restrict__ x2,
    const float* __restrict__ cg, const float* __restrict__ pe,
    float* __restrict__ out, int batch)
{
  const int lane = threadIdx.x & 31;
  const int wave = blockIdx.x * (blockDim.x >> 5) + (threadIdx.x >> 5);
  const int col  = lane & 15;
  const bool hi  = lane >= 16;

  const int groups_per_batch = (FDIM / 16) / TILES_PER_WAVE;  // 8
  const int b = wave / groups_per_batch;
  const int g = wave % groups_per_batch;
  if (b >= batch) return;  // uniform per wave: EXEC stays all-ones for WMMA

  // ---- Preload A fragments (cg transposed) once per wave; reused over 4 tiles.
  // A 16x4 layout: lanes 0-15 -> row M=lane, K = 4k+0 (.x) / 4k+1 (.y)
  //               lanes 16-31 -> row M=lane-16, K = 4k+2 (.x) / 4k+3 (.y)
  v2f A0[NK], A1[NK];
#pragma unroll
  for (int k = 0; k < NK; ++k) {
    const int kk0 = 4 * k + (hi ? 2 : 0);
    const int kk1 = kk0 + 1;
    const int n0 = col;        // tile0 rows n = 0..15
    const int n1 = 16 + col;   // tile1 rows n = 16..31 (valid if < 25)
    float a00 = 0.f, a01 = 0.f, a10 = 0.f, a11 = 0.f;
    if (kk0 < KDIM)              a00 = cg[kk0 * NDIM + n0];
    if (kk1 < KDIM)              a01 = cg[kk1 * NDIM + n0];
    if (kk0 < KDIM && n1 < NDIM) a10 = cg[kk0 * NDIM + n1];
    if (kk1 < KDIM && n1 < NDIM) a11 = cg[kk1 * NDIM + n1];
    v2f t0; t0.x = a00; t0.y = a01; A0[k] = t0;
    v2f t1; t1.x = a10; t1.y = a11; A1[k] = t1;
  }

  const float pe0 = pe[0];
  const float pe1 = pe[1];

  for (int it = 0; it < TILES_PER_WAVE; ++it) {
    const int f = (g * TILES_PER_WAVE + it) * 16 + col;
    const float* x1p = x1 + (size_t)b * L1DIM * FDIM + f;
    const float* x2p = x2 + (size_t)b * L2DIM * FDIM + f;

    float xv[L1DIM], yv[L2DIM];
#pragma unroll
    for (int l = 0; l < L1DIM; ++l) xv[l] = x1p[l * FDIM];
#pragma unroll
    for (int m = 0; m < L2DIM; ++m) yv[m] = x2p[m * FDIM];

    v8f C0 = {};  // rows n = 0..15
    v8f C1 = {};  // rows n = 16..31 (only 16..24 stored)

#pragma unroll
    for (int k = 0; k < NK; ++k) {
      // outer products for this K group (compile-time tail guard)
      float p0 = 0.f, p1 = 0.f, p2 = 0.f, p3 = 0.f;
      if (4 * k + 0 < KDIM) p0 = xv[(4 * k + 0) / 9] * yv[(4 * k + 0) % 9];
      if (4 * k + 1 < KDIM) p1 = xv[(4 * k + 1) / 9] * yv[(4 * k + 1) % 9];
      if (4 * k + 2 < KDIM) p2 = xv[(4 * k + 2) / 9] * yv[(4 * k + 2) % 9];
      if (4 * k + 3 < KDIM) p3 = xv[(4 * k + 3) / 9] * yv[(4 * k + 3) % 9];
      // B 4x16 layout: VGPR0 lanes0-15 K=4k+0 / lanes16-31 K=4k+2; VGPR1 K=4k+1/4k+3
      v2f Bf;
      Bf.x = hi ? p2 : p0;
      Bf.y = hi ? p3 : p1;
      C0 = __builtin_amdgcn_wmma_f32_16x16x4_f32(false, A0[k], false, Bf,
                                                 (short)0, C0, false, false);
      C1 = __builtin_amdgcn_wmma_f32_16x16x4_f32(false, A1[k], false, Bf,
                                                 (short)0, C1, false, false);
    }

    // ---- Store both parity planes.
    // D layout: VGPR r -> n=r (lanes 0-15) / n=r+8 (lanes 16-31), column f.
    float* op = out + (size_t)b * 2 * NDIM * FDIM + f;
#pragma unroll
    for (int r = 0; r < 8; ++r) {
      const int n0 = hi ? (r + 8) : r;
      op[(size_t)n0 * FDIM]          = pe0 * C0[r];
      op[(size_t)(NDIM + n0) * FDIM] = pe1 * C0[r];
      const int n1 = 16 + n0;
      if (n1 < NDIM) {
        op[(size_t)n1 * FDIM]          = pe0 * C1[r];
        op[(size_t)(NDIM + n1) * FDIM] = pe1 * C1[r];
      }
    }
  }
}

extern "C" void kernel_launch(void* const* d_in, const int* in_sizes, int n_in,
                              void* d_out, int out_size, void* d_ws, size_t ws_size,
                              hipStream_t stream) {
  const float* x1 = (const float*)d_in[0];  // (B,1,9,512)
  const float* x2 = (const float*)d_in[1];  // (B,1,9,512)
  const float* cg = (const float*)d_in[2];  // (9,9,25) == (81,25)
  const float* pe = (const float*)d_in[3];  // (2,1)
  float* out = (float*)d_out;               // (B,2,25,512)

  const int batch = in_sizes[0] / (L1DIM * FDIM);
  // 8 waves/block; each wave covers 4 feature tiles -> one block per batch row
  const int groups_per_batch = (FDIM / 16) / TILES_PER_WAVE;  // 8
  const int waves = batch * groups_per_batch;
  const int blocks = (waves + 7) / 8;
  cg_tp_wmma<<<blocks, 256, 0, stream>>>(x1, x2, cg, pe, out, batch);
}